// QNetwork_90288802496576
// MI455X (gfx1250) — compile-verified
//
#include <hip/hip_runtime.h>
#include <hip/hip_bf16.h>
#include <math.h>

// Problem constants (from reference)
#define BOXN   4096
#define XYN    4
#define H1N    100
#define DN     4196      // BOX + H1
#define NUMR   4096      // rows of x
#define ROWLEN 4100      // BOX + XY

typedef __attribute__((ext_vector_type(2))) float v2f;
typedef __attribute__((ext_vector_type(8))) float v8f;

// workspace layout (float offsets)
#define WS_DCOL 0        // 4100 column sums of x
#define WS_SCOL 4100     // 4100 masked column sums of x
#define WS_DVEC 8200     // 4196  d vector
#define WS_TSV  12396    // 4196  tanh(s) vector
#define WS_AVEC 16592    // 4196  action row
// total 20788 floats = 83.2 KB

// ---------------------------------------------------------------- K0: init
__global__ void k0_init(float* __restrict__ ws, float* __restrict__ out,
                        const float* __restrict__ b4) {
  int i = blockIdx.x * blockDim.x + threadIdx.x;
  if (i < 8200) ws[i] = 0.0f;          // zero dcol + scol (atomic targets)
  if (i == 0)  out[0] = b4[0];         // seed scalar accumulator with bias
}

// ------------------------------------------- K1: WMMA f32 column sums of x
// One wave per (16-col strip, 512-row chunk). B tile = x[r..r+3, col0..col0+15]
// (4x16 f32, lanes 0-15 hold K=0/1, lanes 16-31 hold K=2/3 -> 64B coalesced
// runs). A = broadcast ones for plain sum, broadcast mask for masked sum.
// Every row of D equals the 16 column partial sums (A rows identical).
__global__ void __launch_bounds__(32)
k1_colsum_wmma(const float* __restrict__ x, const int* __restrict__ c,
               float* __restrict__ ws) {
  float* dcol = ws + WS_DCOL;
  float* scol = ws + WS_SCOL;
  const int strip = blockIdx.x;            // 0..255  -> cols strip*16..+15
  const int chunk = blockIdx.y;            // 0..7    -> rows chunk*512..+511
  const int lane  = threadIdx.x;           // 0..31 (one full wave, EXEC all 1s)
  const int col0  = strip << 4;
  const int n     = lane & 15;
  const int kh    = (lane >> 4) << 1;      // 0 for lanes 0-15, 2 for 16-31

  v8f accd = {};
  v8f accs = {};
  v2f aones; aones[0] = 1.0f; aones[1] = 1.0f;

  const int row0 = chunk << 9;             // *512
  for (int r = row0; r < row0 + 512; r += 4) {
    const size_t base = (size_t)(r + kh) * ROWLEN + col0 + n;
    v2f b;
    b[0] = x[base];                        // row r+kh
    b[1] = x[base + ROWLEN];               // row r+kh+1
    v2f am;                                // A[m,k] = mask[r+k], same for all m
    am[0] = (c[r + kh]     == 1) ? 1.0f : 0.0f;
    am[1] = (c[r + kh + 1] == 1) ? 1.0f : 0.0f;
    accd = __builtin_amdgcn_wmma_f32_16x16x4_f32(false, aones, false, b,
                                                 (short)0, accd, false, false);
    accs = __builtin_amdgcn_wmma_f32_16x16x4_f32(false, am,    false, b,
                                                 (short)0, accs, false, false);
  }
  // D VGPR0, lanes 0-15 = row M=0, N=lane -> the 16 column sums
  if (lane < 16) {
    atomicAdd(&dcol[col0 + lane], accd[0]);
    atomicAdd(&scol[col0 + lane], accs[0]);
  }
}

// ---------------------- K2: tail (xy-derived) entries, single 128-thread block
// d[4096+j] = dx4 . l1_w[j] + 4096*l1_b[j]
// s[4096+j] = sx4 . l1_w[j] +  cnt*l1_b[j]   (then tanh)
// a[4096+j] = x[a,4096:] . l1_w[j] + l1_b[j]
__global__ void __launch_bounds__(128)
k2_tails(const float* __restrict__ x, const int* __restrict__ c,
         const int* __restrict__ aidx,
         const float* __restrict__ l1w, const float* __restrict__ l1b,
         float* __restrict__ ws) {
  __shared__ float red[9][128];
  const int tid = threadIdx.x;
  float dx0=0,dx1=0,dx2=0,dx3=0, sx0=0,sx1=0,sx2=0,sx3=0, cnt=0;
  for (int i = tid; i < NUMR; i += 128) {
    const float4 t = *reinterpret_cast<const float4*>(x + (size_t)i * ROWLEN + BOXN);
    const float m = (c[i] == 1) ? 1.0f : 0.0f;
    dx0 += t.x;   dx1 += t.y;   dx2 += t.z;   dx3 += t.w;
    sx0 += m*t.x; sx1 += m*t.y; sx2 += m*t.z; sx3 += m*t.w;
    cnt += m;
  }
  red[0][tid]=dx0; red[1][tid]=dx1; red[2][tid]=dx2; red[3][tid]=dx3;
  red[4][tid]=sx0; red[5][tid]=sx1; red[6][tid]=sx2; red[7][tid]=sx3;
  red[8][tid]=cnt;
  __syncthreads();
  for (int s = 64; s > 0; s >>= 1) {
    if (tid < s)
      for (int k = 0; k < 9; ++k) red[k][tid] += red[k][tid + s];
    __syncthreads();
  }
  if (tid < H1N) {
    const int arow = aidx[0];
    const float w0 = l1w[tid*4+0], w1v = l1w[tid*4+1];
    const float w2v = l1w[tid*4+2], w3v = l1w[tid*4+3];
    const float b  = l1b[tid];
    const float dt = red[0][0]*w0 + red[1][0]*w1v + red[2][0]*w2v + red[3][0]*w3v
                   + (float)NUMR * b;
    const float st = red[4][0]*w0 + red[5][0]*w1v + red[6][0]*w2v + red[7][0]*w3v
                   + red[8][0] * b;
    const float4 xa = *reinterpret_cast<const float4*>(x + (size_t)arow * ROWLEN + BOXN);
    const float at = xa.x*w0 + xa.y*w1v + xa.z*w2v + xa.w*w3v + b;
    ws[WS_DVEC + BOXN + tid] = dt;
    ws[WS_TSV  + BOXN + tid] = tanhf(st);
    ws[WS_AVEC + BOXN + tid] = at;
  }
}

// ------------------- K3: assemble main 4096 entries of d, tanh(s), action
__global__ void k3_finalize(const float* __restrict__ x, const int* __restrict__ aidx,
                            float* __restrict__ ws) {
  const int arow = aidx[0];
  for (int i = blockIdx.x * blockDim.x + threadIdx.x; i < BOXN;
       i += gridDim.x * blockDim.x) {
    ws[WS_DVEC + i] = ws[WS_DCOL + i];
    ws[WS_TSV  + i] = tanhf(ws[WS_SCOL + i]);
    ws[WS_AVEC + i] = x[(size_t)arow * ROWLEN + i];
  }
}

// --------- K4: fused dual GEMV + dot reduction (bandwidth-dominant: 140 MB)
// out += sum_r action[r]*(w2[r,:].d - w3[r,:].tanh(s)) + w1[r]*action[r]
__global__ void __launch_bounds__(256)
k4_gemv(const float* __restrict__ w2, const float* __restrict__ w3,
        const float* __restrict__ w1, const float* __restrict__ ws,
        float* __restrict__ out) {
  __shared__ __align__(16) float s_d[DN];
  __shared__ __align__(16) float s_t[DN];
  const int tid = threadIdx.x;
  for (int i = tid; i < DN; i += 256) {
    s_d[i] = ws[WS_DVEC + i];
    s_t[i] = ws[WS_TSV + i];
  }
  __syncthreads();
  const float4* s_d4 = reinterpret_cast<const float4*>(s_d);
  const float4* s_t4 = reinterpret_cast<const float4*>(s_t);

  const int lane = tid & 31;
  const int wave = (blockIdx.x << 3) + (tid >> 5);
  const int nw   = gridDim.x << 3;
  const int NQ   = DN / 4;                 // 1049 float4 per row (16B-aligned)

  for (int r = wave; r < DN; r += nw) {
    const float4* w2r = reinterpret_cast<const float4*>(w2 + (size_t)r * DN);
    const float4* w3r = reinterpret_cast<const float4*>(w3 + (size_t)r * DN);
    float u = 0.0f, v = 0.0f;
    for (int q = lane; q < NQ; q += 32) {  // 512B contiguous per wave per load
      const float4 a2 = w2r[q]; const float4 dd = s_d4[q];
      u += a2.x*dd.x + a2.y*dd.y + a2.z*dd.z + a2.w*dd.w;
      const float4 a3 = w3r[q]; const float4 tt = s_t4[q];
      v += a3.x*tt.x + a3.y*tt.y + a3.z*tt.z + a3.w*tt.w;
    }
    float red = u - v;
    for (int off = 16; off > 0; off >>= 1)
      red += __shfl_xor(red, off, 32);
    if (lane == 0) {
      const float ar = ws[WS_AVEC + r];
      atomicAdd(out, ar * red + w1[r] * ar);
    }
  }
}

// --------------------------------------------------------------- launcher
extern "C" void kernel_launch(void* const* d_in, const int* in_sizes, int n_in,
                              void* d_out, int out_size, void* d_ws, size_t ws_size,
                              hipStream_t stream) {
  const float* x   = (const float*)d_in[0];
  const int*   c   = (const int*)  d_in[1];
  const int*   a   = (const int*)  d_in[2];
  const float* l1w = (const float*)d_in[3];
  const float* l1b = (const float*)d_in[4];
  // d_in[5..7] (l2_w, l2_b, l3_w) feed dead code in the reference: unused.
  const float* w1  = (const float*)d_in[8];
  const float* w2  = (const float*)d_in[9];
  const float* w3  = (const float*)d_in[10];
  const float* b4  = (const float*)d_in[11];
  float* out = (float*)d_out;
  float* ws  = (float*)d_ws;

  k0_init<<<33, 256, 0, stream>>>(ws, out, b4);
  k1_colsum_wmma<<<dim3(256, 8), 32, 0, stream>>>(x, c, ws);
  k2_tails<<<1, 128, 0, stream>>>(x, c, a, l1w, l1b, ws);
  k3_finalize<<<16, 256, 0, stream>>>(x, a, ws);
  k4_gemv<<<256, 256, 0, stream>>>(w2, w3, w1, ws, out);
}